// CentralSplitter_60816736911413
// MI455X (gfx1250) — compile-verified
//
#include <hip/hip_runtime.h>
#include <stdint.h>
#include <stddef.h>

// ---------------- problem constants (match reference) ----------------
#define N_ATOMS   1048576
#define NSPEC     4
#define D_A       128
#define D_B       64
#define NPER      (N_ATOMS / NSPEC)     // 262144
#define CHUNK     1024                  // atoms per block for hist/scatter
#define NCHUNK    (N_ATOMS / CHUNK)     // 1024 blocks
#define T_HEAVY   256                   // 8 waves per block
#define ITEMS_PT  (CHUNK / T_HEAVY)     // 4 atoms per thread

static_assert(T_HEAVY == CHUNK / 4, "one 16B staging unit per thread");

// ---------------- CDNA5 async global->LDS staging --------------------
#if defined(__has_builtin)
#if __has_builtin(__builtin_amdgcn_global_load_async_to_lds_b128)
#define HAVE_ASYNC_LDS 1
#endif
#endif

typedef int   v4i __attribute__((vector_size(16)));
typedef float f4  __attribute__((ext_vector_type(4)));
typedef v4i __attribute__((address_space(1))) *as1_v4i;
typedef v4i __attribute__((address_space(3))) *as3_v4i;

__device__ __forceinline__ f4 nt_load(const f4* p) {
  return __builtin_nontemporal_load(p);
}
__device__ __forceinline__ void nt_store(f4* p, f4 v) {
  __builtin_nontemporal_store(v, p);
}

// Stage CHUNK int32 species ids from global into LDS (16B per thread).
__device__ __forceinline__ void stage_chunk(const int* __restrict__ g,
                                            int* lds, int t) {
#ifdef HAVE_ASYNC_LDS
  __builtin_amdgcn_global_load_async_to_lds_b128(
      (as1_v4i)(v4i*)(g + t * 4),
      (as3_v4i)(v4i*)(lds + t * 4),
      /*offset=*/0, /*cpol=*/0);
  asm volatile("s_wait_asynccnt 0" ::: "memory");
  __syncthreads();
#else
  int4 v = ((const int4*)g)[t];
  ((int4*)lds)[t] = v;
  __syncthreads();
#endif
}

// ---------------- K1: per-chunk species histogram ---------------------
__global__ void __launch_bounds__(T_HEAVY)
k_hist(const int* __restrict__ species, int* __restrict__ hist) {
  __shared__ int sp[CHUNK];
  __shared__ unsigned long long red[T_HEAVY];
  const int t = threadIdx.x;
  const int b = blockIdx.x;

  stage_chunk(species + (size_t)b * CHUNK, sp, t);

  int s0 = sp[t * ITEMS_PT + 0];
  int s1 = sp[t * ITEMS_PT + 1];
  int s2 = sp[t * ITEMS_PT + 2];
  int s3 = sp[t * ITEMS_PT + 3];

  int c0 = (s0 == 0) + (s1 == 0) + (s2 == 0) + (s3 == 0);
  int c1 = (s0 == 1) + (s1 == 1) + (s2 == 1) + (s3 == 1);
  int c2 = (s0 == 2) + (s1 == 2) + (s2 == 2) + (s3 == 2);
  int c3 = (s0 == 3) + (s1 == 3) + (s2 == 3) + (s3 == 3);

  // chunk totals <= 1024: 16-bit fields are safe
  unsigned long long packed =
      (unsigned long long)(unsigned)c0 |
      ((unsigned long long)(unsigned)c1 << 16) |
      ((unsigned long long)(unsigned)c2 << 32) |
      ((unsigned long long)(unsigned)c3 << 48);
  red[t] = packed;
  __syncthreads();
  for (int off = T_HEAVY / 2; off > 0; off >>= 1) {
    if (t < off) red[t] += red[t + off];
    __syncthreads();
  }
  if (t == 0) {
    unsigned long long tot = red[0];
    hist[b * 4 + 0] = (int)( tot        & 0xFFFF);
    hist[b * 4 + 1] = (int)((tot >> 16) & 0xFFFF);
    hist[b * 4 + 2] = (int)((tot >> 32) & 0xFFFF);
    hist[b * 4 + 3] = (int)((tot >> 48) & 0xFFFF);
  }
}

// ---------------- K2: exclusive scan over chunks per species ---------
__global__ void __launch_bounds__(NCHUNK)
k_scan(const int* __restrict__ hist, int* __restrict__ base) {
  __shared__ int sb[NCHUNK];
  const int t = threadIdx.x;   // t == chunk id
  for (int s = 0; s < NSPEC; ++s) {
    int mine = hist[t * 4 + s];
    sb[t] = mine;
    __syncthreads();
    for (int off = 1; off < NCHUNK; off <<= 1) {
      int v = (t >= off) ? sb[t - off] : 0;
      __syncthreads();
      sb[t] += v;
      __syncthreads();
    }
    int excl = sb[t] - mine;
    base[t * 4 + s] = s * NPER + excl;
    __syncthreads();
  }
}

// ---------------- K3: stable ranks + row copy -------------------------
__global__ void __launch_bounds__(T_HEAVY)
k_scatter(const float* __restrict__ fa, const float* __restrict__ fb,
          const int* __restrict__ species, const int* __restrict__ base,
          float* __restrict__ outA, float* __restrict__ outB) {
  __shared__ int sp[CHUNK];
  __shared__ int dstIdx[CHUNK];
  __shared__ unsigned long long sc[T_HEAVY];

  const int t  = threadIdx.x;
  const int b  = blockIdx.x;
  const int cb = b * CHUNK;

  stage_chunk(species + (size_t)cb, sp, t);

  int s0 = sp[t * ITEMS_PT + 0];
  int s1 = sp[t * ITEMS_PT + 1];
  int s2 = sp[t * ITEMS_PT + 2];
  int s3 = sp[t * ITEMS_PT + 3];

  int c0 = (s0 == 0) + (s1 == 0) + (s2 == 0) + (s3 == 0);
  int c1 = (s0 == 1) + (s1 == 1) + (s2 == 1) + (s3 == 1);
  int c2 = (s0 == 2) + (s1 == 2) + (s2 == 2) + (s3 == 2);
  int c3 = (s0 == 3) + (s1 == 3) + (s2 == 3) + (s3 == 3);

  unsigned long long mine =
      (unsigned long long)(unsigned)c0 |
      ((unsigned long long)(unsigned)c1 << 16) |
      ((unsigned long long)(unsigned)c2 << 32) |
      ((unsigned long long)(unsigned)c3 << 48);
  sc[t] = mine;
  __syncthreads();
  for (int off = 1; off < T_HEAVY; off <<= 1) {
    unsigned long long v = (t >= off) ? sc[t - off] : 0ULL;
    __syncthreads();
    sc[t] += v;
    __syncthreads();
  }
  unsigned long long excl = sc[t] - mine;

  // running destination per species for this thread (global dst row index)
  int st0 = base[b * 4 + 0] + (int)( excl        & 0xFFFF);
  int st1 = base[b * 4 + 1] + (int)((excl >> 16) & 0xFFFF);
  int st2 = base[b * 4 + 2] + (int)((excl >> 32) & 0xFFFF);
  int st3 = base[b * 4 + 3] + (int)((excl >> 48) & 0xFFFF);

#pragma unroll
  for (int k = 0; k < ITEMS_PT; ++k) {
    int s = sp[t * ITEMS_PT + k];
    int d;
    if      (s == 0) d = st0++;
    else if (s == 1) d = st1++;
    else if (s == 2) d = st2++;
    else             d = st3++;
    dstIdx[t * ITEMS_PT + k] = d;
  }
  __syncthreads();

  // ---- copy phase: one wave handles 4 atoms per iteration -----------
  const int wave = t >> 5;
  const int lane = t & 31;
  const int half = lane >> 4;     // 0 for lanes 0-15, 1 for lanes 16-31
  const int l16  = lane & 15;

  for (int a = wave * 4; a < CHUNK; a += 32) {
    const int i0 = cb + a;
    const int j0 = dstIdx[a + 0];
    const int j1 = dstIdx[a + 1];
    const int j2 = dstIdx[a + 2];
    const int j3 = dstIdx[a + 3];

    // feat_a: 128 floats/row = 32 f4 = one full wave per row (512B contig)
    const f4* sa = (const f4*)(fa + (size_t)i0 * D_A);
    f4 v0 = nt_load(sa +       lane);
    f4 v1 = nt_load(sa + 32  + lane);
    f4 v2 = nt_load(sa + 64  + lane);
    f4 v3 = nt_load(sa + 96  + lane);

    // feat_b: 64 floats/row; half-wave per row, two adjacent rows per op
    const int  ilo = i0 + half;          // rows a+0 / a+1
    const int  ihi = i0 + 2 + half;      // rows a+2 / a+3
    const int  jlo = half ? j1 : j0;
    const int  jhi = half ? j3 : j2;
    f4 w0 = nt_load((const f4*)(fb + (size_t)ilo * D_B) + l16);
    f4 w1 = nt_load((const f4*)(fb + (size_t)ihi * D_B) + l16);

    if (a + 32 < CHUNK) {
      // stream next iteration's source rows into cache (global_prefetch_b8)
      __builtin_prefetch((const char*)(fa + (size_t)(i0 + 32) * D_A) + lane * 64, 0, 0);
      __builtin_prefetch((const char*)(fb + (size_t)(i0 + 32) * D_B) + lane * 32, 0, 0);
    }

    nt_store((f4*)(outA + (size_t)j0 * D_A) + lane, v0);
    nt_store((f4*)(outA + (size_t)j1 * D_A) + lane, v1);
    nt_store((f4*)(outA + (size_t)j2 * D_A) + lane, v2);
    nt_store((f4*)(outA + (size_t)j3 * D_A) + lane, v3);

    nt_store((f4*)(outB + (size_t)jlo * D_B) + l16, w0);
    nt_store((f4*)(outB + (size_t)jhi * D_B) + l16, w1);
  }
}

// ---------------- host-side launch ------------------------------------
extern "C" void kernel_launch(void* const* d_in, const int* in_sizes, int n_in,
                              void* d_out, int out_size, void* d_ws, size_t ws_size,
                              hipStream_t stream) {
  const float* fa = (const float*)d_in[0];        // feat_a [N, 128] f32
  const float* fb = (const float*)d_in[1];        // feat_b [N, 64]  f32
  const int*   sp = (const int*)  d_in[2];        // central_species [N] i32

  float* outA = (float*)d_out;                    // [S, N/S, 128] flat
  float* outB = outA + (size_t)N_ATOMS * D_A;     // [S, N/S, 64]  flat

  int* hist = (int*)d_ws;                         // NCHUNK*4 ints
  int* base = hist + NCHUNK * 4;                  // NCHUNK*4 ints

  k_hist   <<<NCHUNK, T_HEAVY, 0, stream>>>(sp, hist);
  k_scan   <<<1,      NCHUNK,  0, stream>>>(hist, base);
  k_scatter<<<NCHUNK, T_HEAVY, 0, stream>>>(fa, fb, sp, base, outA, outB);
}